// VectorQuantizer_28166395528103
// MI455X (gfx1250) — compile-verified
//
#include <hip/hip_runtime.h>
#include <hip/hip_bf16.h>

typedef __attribute__((ext_vector_type(2))) float v2f;
typedef __attribute__((ext_vector_type(8))) float v8f;
typedef __attribute__((ext_vector_type(4))) int   v4i;

#define DECAY_F 0.99f
#define OMD_F   0.01f   /* 1 - DECAY */
#define EPS_F   1e-6f

constexpr int Bc = 32, Lc = 1024, Dc = 512, Kc = 8192;
constexpr int Nc = Bc * Lc;                       // 32768 rows

// ---- output layout (floats, concatenated in reference return order) ----
constexpr long long OUT_ZQ   = 0;                               // N*D
constexpr long long OUT_IDX  = (long long)Nc * Dc;              // N
constexpr long long OUT_LOSS = OUT_IDX + Nc;                    // 1
constexpr long long OUT_CB   = OUT_LOSS + 1;                    // K*D
constexpr long long OUT_CS   = OUT_CB + (long long)Kc * Dc;     // K
constexpr long long OUT_EMA  = OUT_CS + Kc;                     // K*D

// ---- workspace layout (floats) ----
constexpr int WS_CBIAS = 0;             // K floats: -0.5*||c||^2
constexpr int WS_IDX   = Kc;            // N ints
constexpr int WS_LOSS  = Kc + Nc;       // 1 float
constexpr int WS_N     = WS_LOSS + 1;   // 1 float

// ---- CDNA5 async global->LDS staging (guarded; falls back to manual copy) ----
#if __has_builtin(__builtin_amdgcn_global_load_async_to_lds_b128) && \
    __has_builtin(__builtin_amdgcn_s_wait_asynccnt)
#define VQ_USE_ASYNC 1
typedef __attribute__((address_space(1))) v4i* vq_gptr;
typedef __attribute__((address_space(3))) v4i* vq_lptr;
#define VQ_ASYNC_CP_F4(dstLds, srcGlb)                                        \
  __builtin_amdgcn_global_load_async_to_lds_b128(                             \
      (vq_gptr)(srcGlb), (vq_lptr)(dstLds), 0, 0)
#else
#define VQ_USE_ASYNC 0
#endif

// ---------------------------------------------------------------------------
// Kernel 1: per-code bias = -0.5*||c||^2. One wave (32 lanes) per codebook row.
// (argmin of distance == argmax of x.c - 0.5*||c||^2)
// ---------------------------------------------------------------------------
__global__ void vq_cbbias(const float* __restrict__ cb, float* __restrict__ cbias) {
  int warp = (int)((blockIdx.x * blockDim.x + threadIdx.x) >> 5);
  int lane = threadIdx.x & 31;
  if (warp >= Kc) return;
  const float4* row = (const float4*)(cb + (long long)warp * Dc);
  float s = 0.f;
  for (int j = 0; j < Dc / 128; ++j) {            // 4 float4 per lane
    float4 v = row[lane + j * 32];
    s += v.x * v.x + v.y * v.y + v.z * v.z + v.w * v.w;
  }
  for (int m = 16; m >= 1; m >>= 1) s += __shfl_xor(s, m, 32);
  if (lane == 0) cbias[warp] = -0.5f * s;
}

// ---------------------------------------------------------------------------
// Kernel 2: init output EMA regions with decay*old, zero scalar accumulators.
// ---------------------------------------------------------------------------
__global__ void vq_init(const float* __restrict__ ema_w, const float* __restrict__ csz,
                        float* __restrict__ out, float* __restrict__ ws) {
  long long i = (long long)blockIdx.x * blockDim.x + threadIdx.x;
  if (i < (long long)Kc * Dc) out[OUT_EMA + i] = DECAY_F * ema_w[i];
  if (i < Kc)                 out[OUT_CS + i]  = DECAY_F * csz[i];
  if (i == 0) { ws[WS_LOSS] = 0.f; ws[WS_N] = 0.f; }
}

// ---------------------------------------------------------------------------
// Kernel 3: WMMA argmax. 64 rows/block (4 waves x 16-row tiles), 32 codes per
// pass with TWO accumulators per wave so the A fragment is reused (cuts LDS
// operand traffic per WMMA). Bias -0.5||c||^2 enters via the WMMA C operand.
// Tiles staged via GLOBAL_LOAD_ASYNC_TO_LDS_B128 when available.
// ---------------------------------------------------------------------------
#define ROWS   64
#define CCOLS  32
#define DCHUNK 128
#define XSTR   132   /* 128 + 4 pad: float4-aligned, LDS bank-conflict free */

__global__ __launch_bounds__(128)
void vq_argmax_wmma(const float* __restrict__ ze, const float* __restrict__ cb,
                    const float* __restrict__ cbias, int* __restrict__ idx_ws) {
  __shared__ float Xs[ROWS * XSTR];    // 33792 B
  __shared__ float Cs[CCOLS * XSTR];   // 16896 B
  const int tid  = threadIdx.x;
  const int lane = tid & 31;
  const int wave = tid >> 5;
  const int hi   = lane >> 4;          // which 16-lane half
  const int lm   = lane & 15;          // column-in-tile / row-M-in-tile
  const long long rowbase = (long long)blockIdx.x * ROWS;

  float maxv[8];
  int   maxi[8];
#pragma unroll
  for (int r = 0; r < 8; ++r) { maxv[r] = -3.4e38f; maxi[r] = 0; }

  for (int cc = 0; cc < Kc / CCOLS; ++cc) {       // 256 passes of 32 codes
    const float bia0 = cbias[cc * CCOLS + lm];
    const float bia1 = cbias[cc * CCOLS + 16 + lm];
    v8f acc0, acc1;
#pragma unroll
    for (int r = 0; r < 8; ++r) { acc0[r] = bia0; acc1[r] = bia1; }

    for (int dcb = 0; dcb < Dc; dcb += DCHUNK) {
      __syncthreads();                            // protect LDS reuse
#if VQ_USE_ASYNC
      for (int q = tid; q < ROWS * (DCHUNK / 4); q += 128) {
        int r = q >> 5, c4 = q & 31;
        VQ_ASYNC_CP_F4(Xs + r * XSTR + c4 * 4,
                       ze + (rowbase + r) * Dc + dcb + c4 * 4);
      }
      for (int q = tid; q < CCOLS * (DCHUNK / 4); q += 128) {
        int r = q >> 5, c4 = q & 31;
        VQ_ASYNC_CP_F4(Cs + r * XSTR + c4 * 4,
                       cb + (long long)(cc * CCOLS + r) * Dc + dcb + c4 * 4);
      }
      __builtin_amdgcn_s_wait_asynccnt(0);
#else
      for (int q = tid; q < ROWS * (DCHUNK / 4); q += 128) {
        int r = q >> 5, c4 = q & 31;
        float4 v = *(const float4*)(ze + (rowbase + r) * Dc + dcb + c4 * 4);
        *(float4*)(Xs + r * XSTR + c4 * 4) = v;
      }
      for (int q = tid; q < CCOLS * (DCHUNK / 4); q += 128) {
        int r = q >> 5, c4 = q & 31;
        float4 v = *(const float4*)(cb + (long long)(cc * CCOLS + r) * Dc + dcb + c4 * 4);
        *(float4*)(Cs + r * XSTR + c4 * 4) = v;
      }
#endif
      __syncthreads();

      // ISA 16x4-f32 A/B layout: lane<16 -> K pair (k0,k0+1); lane>=16 -> (k0+2,k0+3)
      const float* xrow = Xs + (wave * 16 + lm) * XSTR + (hi << 1);
      const float* crow = Cs + lm * XSTR + (hi << 1);
#pragma unroll
      for (int ks = 0; ks < DCHUNK / 4; ++ks) {   // 32 k-steps, 64 WMMAs
        v2f a  = *(const v2f*)(xrow + ks * 4);
        v2f b0 = *(const v2f*)(crow + ks * 4);
        v2f b1 = *(const v2f*)(crow + 16 * XSTR + ks * 4);
        acc0 = __builtin_amdgcn_wmma_f32_16x16x4_f32(
            false, a, false, b0, (short)0, acc0, false, false);
        acc1 = __builtin_amdgcn_wmma_f32_16x16x4_f32(
            false, a, false, b1, (short)0, acc1, false, false);
      }
    }
    // fold both 16-column tiles; lower columns first so strict > keeps
    // the earliest (lowest-index) maximum, matching argmin tie-breaking.
    const int c0 = cc * CCOLS + lm;
#pragma unroll
    for (int r = 0; r < 8; ++r) {
      float s0 = acc0[r];
      if (s0 > maxv[r]) { maxv[r] = s0; maxi[r] = c0; }
      float s1 = acc1[r];
      if (s1 > maxv[r]) { maxv[r] = s1; maxi[r] = c0 + 16; }
    }
  }

  // reduce across the 16 lanes that share a row; tie-break toward lower index
#pragma unroll
  for (int r = 0; r < 8; ++r) {
    float v = maxv[r];
    int   i = maxi[r];
#pragma unroll
    for (int m = 8; m >= 1; m >>= 1) {
      float ov = __shfl_xor(v, m, 32);
      int   oi = __shfl_xor(i, m, 32);
      if (ov > v || (ov == v && oi < i)) { v = ov; i = oi; }
    }
    if (lm == 0) {
      long long grow = rowbase + wave * 16 + (hi << 3) + r;
      idx_ws[grow] = i;
    }
  }
}

// ---------------------------------------------------------------------------
// Kernel 4: gather z_q, straight-through output, loss partials, EMA scatter.
// One block per row (128 threads x float4 = 512 elements).
// ---------------------------------------------------------------------------
__global__ __launch_bounds__(128)
void vq_gather_scatter(const float* __restrict__ ze, const float* __restrict__ cb,
                       const int* __restrict__ idx_ws, float* __restrict__ out,
                       float* __restrict__ ws) {
  __shared__ float red[4];
  const long long row = blockIdx.x;
  const int tid = threadIdx.x;
  const int k = idx_ws[row];
  const int d = tid * 4;

  float4 x = *(const float4*)(ze + row * Dc + d);
  float4 q = *(const float4*)(cb + (long long)k * Dc + d);
  float4 st;
  st.x = x.x + (q.x - x.x); st.y = x.y + (q.y - x.y);
  st.z = x.z + (q.z - x.z); st.w = x.w + (q.w - x.w);
  *(float4*)(out + OUT_ZQ + row * Dc + d) = st;

  float dx = x.x - q.x, dy = x.y - q.y, dz = x.z - q.z, dw = x.w - q.w;
  float s = dx * dx + dy * dy + dz * dz + dw * dw;

  // EMA scatter: new_ema_w[k] += (1-decay) * x   (region only 4B-aligned -> scalar atomics)
  float* ew = out + OUT_EMA + (long long)k * Dc + d;
  atomicAdd(ew + 0, OMD_F * x.x);
  atomicAdd(ew + 1, OMD_F * x.y);
  atomicAdd(ew + 2, OMD_F * x.z);
  atomicAdd(ew + 3, OMD_F * x.w);

  for (int m = 16; m >= 1; m >>= 1) s += __shfl_xor(s, m, 32);
  if ((tid & 31) == 0) red[tid >> 5] = s;
  __syncthreads();
  if (tid == 0) {
    atomicAdd(ws + WS_LOSS, red[0] + red[1] + red[2] + red[3]);
    atomicAdd(out + OUT_CS + k, OMD_F);      // counts contribution
    out[OUT_IDX + row] = (float)k;           // idx output as float
  }
}

// ---------------------------------------------------------------------------
// Kernel 5: n = sum(new_cluster_size)
// ---------------------------------------------------------------------------
__global__ void vq_sum_cs(const float* __restrict__ out, float* __restrict__ ws) {
  __shared__ float red[256];
  const int tid = threadIdx.x;
  float s = 0.f;
  for (int i = tid; i < Kc; i += 256) s += out[OUT_CS + i];
  red[tid] = s;
  __syncthreads();
  for (int off = 128; off >= 1; off >>= 1) {
    if (tid < off) red[tid] += red[tid + off];
    __syncthreads();
  }
  if (tid == 0) ws[WS_N] = red[0];
}

// ---------------------------------------------------------------------------
// Kernel 6: new_codebook = new_ema_w / smoothed_counts; finalize loss.
// ---------------------------------------------------------------------------
__global__ void vq_finalize(float* __restrict__ out, const float* __restrict__ ws) {
  long long i = (long long)blockIdx.x * blockDim.x + threadIdx.x;
  if (i == 0)
    out[OUT_LOSS] = 1.25f * ws[WS_LOSS] / (float)((long long)Nc * Dc);
  if (i < (long long)Kc * Dc) {
    int k = (int)(i >> 9);                 // D = 512
    float n = ws[WS_N];
    float csk = out[OUT_CS + k];
    float smooth = (csk + EPS_F) / (n + (float)Kc * EPS_F) * n;
    out[OUT_CB + i] = out[OUT_EMA + i] / smooth;
  }
}

// ---------------------------------------------------------------------------
extern "C" void kernel_launch(void* const* d_in, const int* in_sizes, int n_in,
                              void* d_out, int out_size, void* d_ws, size_t ws_size,
                              hipStream_t stream) {
  const float* ze   = (const float*)d_in[0];   // [32,1024,512]
  const float* cb   = (const float*)d_in[1];   // [8192,512]
  const float* csz  = (const float*)d_in[2];   // [8192]
  const float* emaw = (const float*)d_in[3];   // [8192,512]
  float* out = (float*)d_out;
  float* ws  = (float*)d_ws;
  float* cbias  = ws + WS_CBIAS;
  int*   idx_ws = (int*)(ws + WS_IDX);

  const long long KD = (long long)Kc * Dc;

  vq_cbbias<<<Kc / 8, 256, 0, stream>>>(cb, cbias);
  vq_init<<<(int)((KD + 255) / 256), 256, 0, stream>>>(emaw, csz, out, ws);
  vq_argmax_wmma<<<Nc / ROWS, 128, 0, stream>>>(ze, cb, cbias, idx_ws);
  vq_gather_scatter<<<Nc, 128, 0, stream>>>(ze, cb, idx_ws, out, ws);
  vq_sum_cs<<<1, 256, 0, stream>>>(out, ws);
  vq_finalize<<<(int)((KD + 255) / 256), 256, 0, stream>>>(out, ws);
}